// SOINNPLUS_14001593385388
// MI455X (gfx1250) — compile-verified
//
#include <hip/hip_runtime.h>
#include <math.h>

// SOINN+ on MI455X:
//   k1: per-node squared norms (100k waves, shuffle reduce)
//   k2: A[t][j] = ||v0_j||^2 - 2*dot(x_t, v0_j) via V_WMMA_F32_16X16X4_F32
//   k3: single-workgroup sequential 64-step top-2 + dirty-row (copy-on-write)
//       correction; only <=128 rows ever diverge from V0.

#define NNODES 100000
#define DIM    256
#define NSAMP  64
#define MAXDIRTY 128
#define MASKW  3136   // ceil(100000/32)=3125, padded

typedef __attribute__((ext_vector_type(2))) float v2f;
typedef __attribute__((ext_vector_type(8))) float v8f;

__device__ __forceinline__ void top2_merge(float v, int i,
                                           float& d1, int& i1,
                                           float& d2, int& i2) {
  if (i == i1 || i == i2) return;   // dedup (duplicates carry equal values)
  if (v < d1) { d2 = d1; i2 = i1; d1 = v; i1 = i; }
  else if (v < d2) { d2 = v; i2 = i; }
}

// ---------------- kernel 1: node squared norms ----------------
__global__ void soinn_norms(const float* __restrict__ V,
                            float* __restrict__ norm0) {
  const int lane = threadIdx.x & 31;
  const int wv   = threadIdx.x >> 5;
  const int node = blockIdx.x * 8 + wv;
  if (node >= NNODES) return;
  const float* vp = V + (size_t)node * DIM;
  float p = 0.0f;
  for (int i = lane; i < DIM; i += 32) { float x = vp[i]; p = fmaf(x, x, p); }
  for (int off = 16; off; off >>= 1) p += __shfl_xor(p, off);
  if (lane == 0) norm0[node] = p;
}

// ---------------- kernel 2: WMMA f32 GEMM + epilogue ----------------
// grid = 6250 blocks (16 nodes each), 128 threads (4 waves, one 16-sample
// M-tile per wave). A(16x4 f32)=2 VGPRs, B(4x16 f32)=2 VGPRs, C/D=8 VGPRs.
__global__ __launch_bounds__(128) void soinn_gemm(
    const float* __restrict__ data, const float* __restrict__ V,
    const float* __restrict__ norm0, float* __restrict__ Aout) {
  __shared__ float Vt[16 * 260];            // padded pitch: no bank conflicts
  const int tid = threadIdx.x;
  const int n0  = blockIdx.x * 16;

  for (int idx = tid; idx < 16 * DIM; idx += 128) {
    int r = idx >> 8, c = idx & 255;
    Vt[r * 260 + c] = V[(size_t)(n0 + r) * DIM + c];
  }
  __syncthreads();

  const int lane = tid & 31;
  const int wv   = tid >> 5;
  const int half = lane >> 4;      // 0: K pair {0,1}, 1: K pair {2,3}
  const int l16  = lane & 15;      // M (for A) / N (for B) index
  const int m0   = wv * 16;

  v8f acc = {};
  const float* arow = data + (size_t)(m0 + l16) * DIM + 2 * half;
  const int    boff = l16 * 260 + 2 * half;

  for (int k = 0; k < DIM; k += 4) {
    v2f a, b;
    a.x = arow[k];          a.y = arow[k + 1];
    b.x = Vt[boff + k];     b.y = Vt[boff + k + 1];
    acc = __builtin_amdgcn_wmma_f32_16x16x4_f32(
        false, a, false, b, (short)0, acc, false, false);
  }

  const float nrm = norm0[n0 + l16];
#pragma unroll
  for (int i = 0; i < 8; ++i) {
    int t = m0 + i + 8 * half;                  // C/D: VGPR i -> M=i / i+8
    Aout[(size_t)t * NNODES + n0 + l16] = nrm - 2.0f * acc[i];
  }
}

// ---------------- kernel 3: sequential 64-step top-2 ----------------
__global__ __launch_bounds__(1024) void soinn_seq(
    const float* __restrict__ data, const float* __restrict__ V,
    const int* __restrict__ itp, const float* __restrict__ Aout,
    float* __restrict__ dirtyRows, float* __restrict__ out) {
  __shared__ unsigned int mask[MASKW];
  __shared__ float xs[DIM];
  __shared__ float red[DIM];
  __shared__ float wd1[32], wd2[32];
  __shared__ int   wi1[32], wi2[32];
  __shared__ int   dirtyIdx[MAXDIRTY];
  __shared__ int   sNd, sB, sS, sSlotB, sSlotS, sNewB, sNewS;
  __shared__ float sXnorm;

  const int tid  = threadIdx.x;
  const int lane = tid & 31;
  const int wv   = tid >> 5;

  for (int i = tid; i < MASKW; i += 1024) mask[i] = 0u;
  if (tid == 0) sNd = 0;
  __syncthreads();

  const int   it    = itp[0];
  const float eps_b = 1.0f / (float)(it + 2);
  const float eps_n = 1.0f / (float)((it + 1) * 100);

  for (int t = 0; t < NSAMP; ++t) {
    if (tid < DIM) xs[tid] = data[t * DIM + tid];
    __syncthreads();
    if (tid < DIM) red[tid] = xs[tid] * xs[tid];
    __syncthreads();
    for (int s2 = 128; s2 > 0; s2 >>= 1) {      // deterministic tree reduce
      if (tid < s2) red[tid] += red[tid + s2];
      __syncthreads();
    }
    if (tid == 0) sXnorm = red[0];
    __syncthreads();

    const int nd = sNd;
    float d1 = INFINITY, d2 = INFINITY;
    int   i1 = -1, i2 = -1;

    // exact values for dirty rows (excluded from the clean scan by bitmask)
    for (int d = wv; d < nd; d += 32) {
      const float* vp = dirtyRows + d * DIM;
      float p = 0.0f;
      for (int i = lane; i < DIM; i += 32) {
        float v = vp[i];
        p = fmaf(v, v - 2.0f * xs[i], p);       // ||v||^2 - 2 x.v partial
      }
      for (int off = 16; off; off >>= 1) p += __shfl_xor(p, off);
      if (lane == 0) top2_merge(p, dirtyIdx[d], d1, i1, d2, i2);
    }

    // clean scan over A[t][*] (values are dist^2 - ||x||^2)
    const float* arow = Aout + (size_t)t * NNODES;
    for (int j = tid; j < NNODES; j += 1024) {
      if ((mask[j >> 5] >> (j & 31)) & 1u) continue;
      top2_merge(arow[j], j, d1, i1, d2, i2);
    }

    // wave butterfly top-2 reduce
    for (int off = 16; off; off >>= 1) {
      float o1 = __shfl_xor(d1, off); int p1 = __shfl_xor(i1, off);
      float o2 = __shfl_xor(d2, off); int p2 = __shfl_xor(i2, off);
      top2_merge(o1, p1, d1, i1, d2, i2);
      top2_merge(o2, p2, d1, i1, d2, i2);
    }
    if (lane == 0) { wd1[wv] = d1; wi1[wv] = i1; wd2[wv] = d2; wi2[wv] = i2; }
    __syncthreads();

    if (tid < 32) {
      float e1 = wd1[tid], e2 = wd2[tid];
      int   j1 = wi1[tid], j2 = wi2[tid];
      for (int off = 16; off; off >>= 1) {
        float o1 = __shfl_xor(e1, off); int p1 = __shfl_xor(j1, off);
        float o2 = __shfl_xor(e2, off); int p2 = __shfl_xor(j2, off);
        top2_merge(o1, p1, e1, j1, e2, j2);
        top2_merge(o2, p2, e1, j1, e2, j2);
      }
      if (tid == 0) {
        const float xn = sXnorm;
        out[2 * t + 0] = sqrtf(fmaxf(e1 + xn, 0.0f) + 1e-12f);
        out[2 * t + 1] = sqrtf(fmaxf(e2 + xn, 0.0f) + 1e-12f);

        int nd2 = sNd;
        const int b = j1, s = j2;
        int slotB = -1;
        for (int d = 0; d < nd2; ++d) if (dirtyIdx[d] == b) { slotB = d; break; }
        const int newB = (slotB < 0);
        if (newB) { slotB = nd2; dirtyIdx[nd2] = b;
                    mask[b >> 5] |= (1u << (b & 31)); ++nd2; }
        int slotS = -1;
        for (int d = 0; d < nd2; ++d) if (dirtyIdx[d] == s) { slotS = d; break; }
        const int newS = (slotS < 0);
        if (newS) { slotS = nd2; dirtyIdx[nd2] = s;
                    mask[s >> 5] |= (1u << (s & 31)); ++nd2; }
        sNd = nd2; sB = b; sS = s;
        sSlotB = slotB; sSlotS = slotS; sNewB = newB; sNewS = newS;
      }
    }
    __syncthreads();

    // copy-on-write + BMU / sBMU pull (V0 input is never mutated)
    if (tid < DIM) {
      float* rp = dirtyRows + sSlotB * DIM;
      float v = sNewB ? V[(size_t)sB * DIM + tid] : rp[tid];
      rp[tid] = v + eps_b * (xs[tid] - v);
    } else if (tid < 2 * DIM) {
      const int d = tid - DIM;
      float* rp = dirtyRows + sSlotS * DIM;
      float v = sNewS ? V[(size_t)sS * DIM + d] : rp[d];
      rp[d] = v + eps_n * (xs[d] - v);
    }
    __syncthreads();
  }
}

extern "C" void kernel_launch(void* const* d_in, const int* in_sizes, int n_in,
                              void* d_out, int out_size, void* d_ws, size_t ws_size,
                              hipStream_t stream) {
  const float* data = (const float*)d_in[0];   // [64,256] f32
  const float* V    = (const float*)d_in[1];   // [100000,256] f32
  // d_in[2]: labels (unused)
  const int*   itp  = (const int*)d_in[3];     // scalar it
  float* out = (float*)d_out;                  // [64,2] f32

  // workspace layout (floats): A [64*100000] | norm0 [100000] | dirty [128*256]
  float* ws    = (float*)d_ws;
  float* Aout  = ws;
  float* norm0 = ws + (size_t)NSAMP * NNODES;         // +25.6 MB
  float* dirty = norm0 + NNODES;                      // +0.4 MB (128 KB used)

  soinn_norms<<<NNODES / 8, 256, 0, stream>>>(V, norm0);
  soinn_gemm <<<NNODES / 16, 128, 0, stream>>>(data, V, norm0, Aout);
  soinn_seq  <<<1, 1024, 0, stream>>>(data, V, itp, Aout, dirty, out);
}